// CapsuleLayer_18056042512587
// MI455X (gfx1250) — compile-verified
//
#include <hip/hip_runtime.h>
#include <math.h>

// ---------------- problem constants ----------------
#define BATCH 32
#define N_IN  2048
#define D_IN  16
#define N_CAP 64
#define D_CAP 32

// ---------------- tiling ----------------
#define NB 2                 // batch blocks
#define MB 16                // batch rows per block (one WMMA M tile)
#define CN 16                // n values per chunk
#define NC (N_IN / CN)       // 128 chunks
#define WG_THREADS 512       // 16 waves
#define NWAVES 16
#define CAPS_PER_WAVE 4      // 64 capsules / 16 waves
#define TILES_PER_WAVE 8     // 4 capsules x 2 e-tiles (e=32 -> two 16-wide tiles)

// LDS layout (dynamic):
//   u  : MB*N_CAP rows, padded stride 33 floats (bank-conflict avoidance)
//   x  : MB*D_IN floats
//   bb : MB*N_CAP floats (routing logits)
//   cc : MB*N_CAP floats (softmax coupling coeffs)
#define U_STRIDE 33
#define U_FLOATS (MB * N_CAP * U_STRIDE)
#define SMEM_FLOATS (U_FLOATS + MB * D_IN + MB * N_CAP + MB * N_CAP)

typedef __attribute__((ext_vector_type(2))) float v2f;
typedef __attribute__((ext_vector_type(8))) float v8f;

// ---------------- kernels ----------------

__global__ __launch_bounds__(256) void caps_zero_kernel(float* __restrict__ p, int n) {
    int i = blockIdx.x * 256 + threadIdx.x;
    if (i < n) p[i] = 0.0f;
}

// One fused routing phase:
//   per (batch-block, n-chunk) workgroup, loop n in chunk:
//     u = x_n @ W_n + B          (V_WMMA_F32_16X16X4_F32, fp32 tensor path)
//     bb[b,c] = <u[b,c,:], Vsum[b,c,:]>        (logits; Vsum = sum of prior v's)
//     cc = softmax_c(bb)
//     s_acc[b,c,e] += cc[b,c] * u[b,c,e]       (kept in WMMA accumulator regs)
//   write s_acc to per-WG partial buffer.
__global__ __launch_bounds__(WG_THREADS) void caps_phase_kernel(
    const float* __restrict__ x, const float* __restrict__ W,
    const float* __restrict__ Bias, const float* __restrict__ Vsum,
    float* __restrict__ partials)
{
    extern __shared__ float smem[];
    float* u_lds  = smem;
    float* x_lds  = smem + U_FLOATS;
    float* bb_lds = x_lds + MB * D_IN;
    float* cc_lds = bb_lds + MB * N_CAP;

    const int tid   = threadIdx.x;
    const int wave  = tid >> 5;
    const int lane  = tid & 31;
    const int lhalf = lane >> 4;   // 0 or 1 (lane group)
    const int lrow  = lane & 15;

    const int wg     = blockIdx.x;   // 0 .. NB*NC-1
    const int bblk   = wg / NC;
    const int nchunk = wg % NC;
    const int bbase  = bblk * MB;

    // Per-tile bias (same for every M row of a tile): tile j -> capsule c, col e
    float bias_t[TILES_PER_WAVE];
#pragma unroll
    for (int j = 0; j < TILES_PER_WAVE; ++j) {
        int c = wave * CAPS_PER_WAVE + (j >> 1);
        int e = (j & 1) * 16 + lrow;
        bias_t[j] = Bias[c * D_CAP + e];
    }

    // s partial accumulators live in registers, same layout as WMMA D tiles.
    v8f s_acc[TILES_PER_WAVE];
#pragma unroll
    for (int j = 0; j < TILES_PER_WAVE; ++j) s_acc[j] = (v8f){0,0,0,0,0,0,0,0};

    for (int ni = 0; ni < CN; ++ni) {
        const int n = nchunk * CN + ni;

        // ---- stage x[bbase..+16)[n][0..16) into LDS (coalesced) ----
        if (tid < MB * D_IN) {
            int m = tid >> 4, d = tid & 15;
            x_lds[tid] = x[((size_t)(bbase + m) * N_IN + n) * D_IN + d];
        }
        __syncthreads();

        // ---- A fragments (16x4 fp32 per k-step): lane row M=lrow,
        //      VGPR0 holds K = 4k+2*lhalf, VGPR1 holds K = 4k+2*lhalf+1 ----
        v2f a_frag[4];
#pragma unroll
        for (int k = 0; k < 4; ++k) {
            int d0 = 4 * k + 2 * lhalf;
            a_frag[k].x = x_lds[lrow * D_IN + d0];
            a_frag[k].y = x_lds[lrow * D_IN + d0 + 1];
        }

        // ---- GEMM: u tiles (16 batches x 16 e-cols) per capsule ----
        v8f u_tile[TILES_PER_WAVE];
#pragma unroll
        for (int j = 0; j < TILES_PER_WAVE; ++j) {
            const int c  = wave * CAPS_PER_WAVE + (j >> 1);
            const int et = (j & 1);
            const int e  = et * 16 + lrow;
            const float* wbase = W + ((size_t)n * N_CAP + c) * (D_IN * D_CAP);
            v8f acc = (v8f){0,0,0,0,0,0,0,0};
#pragma unroll
            for (int k = 0; k < 4; ++k) {
                int d0 = 4 * k + 2 * lhalf;
                v2f b_frag;
                b_frag.x = wbase[d0 * D_CAP + e];
                b_frag.y = wbase[(d0 + 1) * D_CAP + e];
                acc = __builtin_amdgcn_wmma_f32_16x16x4_f32(
                    /*neg_a=*/false, a_frag[k], /*neg_b=*/false, b_frag,
                    /*c_mod=*/(short)0, acc, /*reuse_a=*/false, /*reuse_b=*/false);
            }
            // bias, stash u, spill tile to padded LDS (D layout: row b = v + 8*lhalf)
#pragma unroll
            for (int v = 0; v < 8; ++v) {
                float uval = acc[v] + bias_t[j];
                acc[v] = uval;
                int bl = v + 8 * lhalf;
                u_lds[(bl * N_CAP + c) * U_STRIDE + e] = uval;
            }
            u_tile[j] = acc;
        }
        __syncthreads();

        // ---- logits bb[b,c] = <u[b,c,:], Vsum[b,c,:]> ; 1024 pairs / 512 thr ----
#pragma unroll
        for (int p = 0; p < 2; ++p) {
            int pair = tid + p * WG_THREADS;
            int bl = pair >> 6;
            int c  = pair & 63;
            const float* vs = Vsum + ((size_t)(bbase + bl) * N_CAP + c) * D_CAP;
            const float* ur = u_lds + (bl * N_CAP + c) * U_STRIDE;
            float acc = 0.0f;
            for (int e = 0; e < D_CAP; ++e) acc += ur[e] * vs[e];
            bb_lds[pair] = acc;
        }
        __syncthreads();

        // ---- softmax over the 64 capsules, one thread per batch row ----
        if (tid < MB) {
            const float* row = bb_lds + tid * N_CAP;
            float* crow = cc_lds + tid * N_CAP;
            float mx = -3.0e38f;
            for (int c = 0; c < N_CAP; ++c) mx = fmaxf(mx, row[c]);
            float sum = 0.0f;
            for (int c = 0; c < N_CAP; ++c) { float ev = __expf(row[c] - mx); crow[c] = ev; sum += ev; }
            float inv = 1.0f / sum;
            for (int c = 0; c < N_CAP; ++c) crow[c] *= inv;
        }
        __syncthreads();

        // ---- s_acc += cc * u   (registers; cc broadcast from LDS) ----
#pragma unroll
        for (int j = 0; j < TILES_PER_WAVE; ++j) {
            const int c = wave * CAPS_PER_WAVE + (j >> 1);
#pragma unroll
            for (int v = 0; v < 8; ++v) {
                int bl = v + 8 * lhalf;
                s_acc[j][v] += cc_lds[bl * N_CAP + c] * u_tile[j][v];
            }
        }
        __syncthreads();   // protect x_lds/u_lds before next n
    }

    // ---- write per-workgroup s partials: partials[wg][b_local][c][e] ----
#pragma unroll
    for (int j = 0; j < TILES_PER_WAVE; ++j) {
        const int c = wave * CAPS_PER_WAVE + (j >> 1);
        const int e = (j & 1) * 16 + lrow;
#pragma unroll
        for (int v = 0; v < 8; ++v) {
            int bl = v + 8 * lhalf;
            partials[(((size_t)wg * MB + bl) * N_CAP + c) * D_CAP + e] = s_acc[j][v];
        }
    }
}

// Reduce partials over n-chunks, squash, accumulate into Vsum, emit v.
// One 32-thread (single wave) block per (b, c); lane = e.
__global__ __launch_bounds__(32) void caps_reduce_squash(
    const float* __restrict__ partials, float* __restrict__ Vsum,
    float* __restrict__ out)
{
    const int pairid = blockIdx.x;        // b*64 + c
    const int b = pairid >> 6;
    const int c = pairid & 63;
    const int e = threadIdx.x;
    const int bblk = b >> 4, bl = b & 15;

    float s = 0.0f;
    for (int nc = 0; nc < NC; ++nc) {
        int wg = bblk * NC + nc;
        s += partials[(((size_t)wg * MB + bl) * N_CAP + c) * D_CAP + e];
    }
    // ||s||^2 across the 32-lane capsule vector (wave32 butterfly)
    float sq = s * s;
#pragma unroll
    for (int off = 16; off > 0; off >>= 1) sq += __shfl_xor(sq, off, 32);
    float scale = sq / (1.0f + sq) * rsqrtf(sq + 1e-9f);
    float v = scale * s;

    size_t idx = ((size_t)b * N_CAP + c) * D_CAP + e;
    Vsum[idx] += v;     // logits at next phase: u . Vsum
    out[idx] = v;       // last phase's write is the final answer
}

// ---------------- host launcher ----------------
extern "C" void kernel_launch(void* const* d_in, const int* in_sizes, int n_in,
                              void* d_out, int out_size, void* d_ws, size_t ws_size,
                              hipStream_t stream) {
    const float* x    = (const float*)d_in[0];
    const float* W    = (const float*)d_in[1];
    const float* Bias = (const float*)d_in[2];

    float* Vsum     = (float*)d_ws;                                   // 256 KB
    float* partials = (float*)((char*)d_ws +
                       (size_t)BATCH * N_CAP * D_CAP * sizeof(float)); // 32 MB

    const size_t smem = SMEM_FLOATS * sizeof(float);  // ~141 KB dynamic LDS
    static bool attr_set = false;  // idempotent; same effect every call
    (void)attr_set;
    hipFuncSetAttribute((const void*)caps_phase_kernel,
                        hipFuncAttributeMaxDynamicSharedMemorySize, (int)smem);

    // Vsum = 0  (phase 0 logits are then 0 -> uniform softmax, matching b0=0)
    caps_zero_kernel<<<(BATCH * N_CAP * D_CAP + 255) / 256, 256, 0, stream>>>(
        Vsum, BATCH * N_CAP * D_CAP);

    for (int phase = 0; phase < 3; ++phase) {
        caps_phase_kernel<<<NB * NC, WG_THREADS, smem, stream>>>(
            x, W, Bias, Vsum, partials);
        caps_reduce_squash<<<BATCH * N_CAP, 32, 0, stream>>>(
            partials, Vsum, (float*)d_out);
    }
}